// MyLinear_34522947125845
// MI455X (gfx1250) — compile-verified
//
#include <hip/hip_runtime.h>
#include <stdint.h>

typedef unsigned int u32;
typedef unsigned long long u64;

// 256 threads; block tile 32(M) x 64(N); thread tile 2x4; k-chunks of 32,
// raw tiles double-buffered via gfx1250 async-to-LDS, then decoded ONCE per
// element into packed feature planes in LDS (decode redundancy eliminated).
#define BM 32
#define BN 64
#define KC 32
#define TM 2
#define TN 4

// ---- LDS layout (dword offsets) ----
#define RAWX   0        // raw x tile, 2 buffers x (32 rows * 32 k) = 2x1024
#define RAWW   2048     // raw w tile, 2 buffers x (64 rows * 32 k) = 2x2048
#define FX0    6144     // x feature plane0: U | Eenc<<21 | sign   (stride 34)
#define FX1    7232     // x feature plane1: V5
#define FW0    8320     // w feature plane0: mb                    (stride 68)
#define FW1    10496    // w feature plane1: Q | Eenc<<21 | sign
#define LDS_DW 12672    // 50688 bytes
#define XFS    34       // padded stride: even (b64-aligned), conflict-free
#define WFS    68       // padded stride: 16B-aligned (b128), conflict-free

// ---- CDNA5 async-tensor path (gfx1250) ----
__device__ __forceinline__ void async_ld_b32(u32 lds_addr, u32 voff, u64 base) {
  // GVS mode: mem = SADDR(64b sgpr pair) + VADDR(32b vgpr byte offset)
  asm volatile("global_load_async_to_lds_b32 %0, %1, %2"
               :: "v"(lds_addr), "v"(voff), "s"(base)
               : "memory");
}
__device__ __forceinline__ void wait_async_12() {
  // async loads complete in order: cnt<=12 => all of the *current* chunk done
  asm volatile("s_wait_asynccnt 12" ::: "memory");
}
__device__ __forceinline__ void wait_async_0() {
  asm volatile("s_wait_asynccnt 0" ::: "memory");
}

__global__ void __launch_bounds__(256, 1)
axc_linear_kernel(const float* __restrict__ xg, const float* __restrict__ wg,
                  const float* __restrict__ bg, float* __restrict__ outg,
                  int K_, int N_)
{
  __shared__ u32 lds[LDS_DW];

  const int tid = threadIdx.x;
  const int tx  = tid & 15;     // 16 col-groups * TN = 64 cols
  const int ty  = tid >> 4;     // 16 row-groups * TM = 32 rows
  const int kkL = tid & 31;     // k-lane for load/decode (coalesced)
  const int g   = tid >> 5;     // 0..7

  const u64 xbase = (u64)(uintptr_t)(xg + (size_t)blockIdx.y * BM * K_);
  const u64 wbase = (u64)(uintptr_t)(wg + (size_t)blockIdx.x * BN * K_);
  const u32 lds0  = (u32)(uintptr_t)(&lds[0]);   // LDS byte address base

  const u32 TINY = __float_as_uint(1.0e-36f);

  // ---- async issue of one raw chunk (12 instructions / thread) ----
  auto issue = [&](int rb, int kc) {
#pragma unroll
    for (int i = 0; i < 4; ++i) {                 // 32 x-rows
      int row = g + 8 * i;
      async_ld_b32(lds0 + (u32)((RAWX + rb * 1024 + row * 32 + kkL) * 4),
                   (u32)((row * K_ + kc + kkL) * 4), xbase);
    }
#pragma unroll
    for (int i = 0; i < 8; ++i) {                 // 64 w-rows
      int row = g + 8 * i;
      async_ld_b32(lds0 + (u32)((RAWW + rb * 2048 + row * 32 + kkL) * 4),
                   (u32)((row * K_ + kc + kkL) * 4), wbase);
    }
  };

  float acc[TM][TN];
#pragma unroll
  for (int r = 0; r < TM; ++r)
#pragma unroll
    for (int c = 0; c < TN; ++c) acc[r][c] = 0.0f;

  issue(0, 0);

  int rb = 0;
  for (int kc = 0; kc < K_; kc += KC) {
    const bool has_next = (kc + KC) < K_;
    if (has_next) { issue(rb ^ 1, kc + KC); wait_async_12(); }
    else          { wait_async_0(); }
    __syncthreads();                    // raw chunk visible to all waves

    // ---- decode ONCE per element into packed feature planes ----
    // AxC24 collapse:  m = U*mb + V5*Q
    //   U  = ma & 0x111111            (odd bits of nibbles)
    //   V5 = (ma & 0xEEEEEE) << 4     (upper nibble bits, x32 pre-baked)
    //   Q  = (mb >> 3) & 0x111111     (bit3 of nibbles)
    // Eenc = z ? 0 : E+256  ->  e = clamp(xEe+wEe-639) == clamp(ea+eb-127),
    // and any zero-class operand forces e==0 and m==0  => product = +-0.0f.
#pragma unroll
    for (int i = 0; i < 4; ++i) {
      int row = 4 * g + i;
      u32 b  = lds[RAWX + rb * 1024 + row * 32 + kkL];
      bool z = (b & 0x7FFFFFFFu) < TINY;
      u32 ma = z ? 0u : ((b & 0x7FFFFFu) | 0x800000u);
      u32 Ee = z ? 0u : (((b >> 23) & 0xFFu) | 0x100u);
      lds[FX0 + kkL * XFS + row] =
          (ma & 0x111111u) | (Ee << 21) | (b & 0x80000000u);
      lds[FX1 + kkL * XFS + row] = (ma & 0xEEEEEEu) << 4;
    }
#pragma unroll
    for (int i = 0; i < 8; ++i) {
      int row = 8 * g + i;
      u32 b  = lds[RAWW + rb * 2048 + row * 32 + kkL];
      bool z = (b & 0x7FFFFFFFu) < TINY;
      u32 mb = z ? 0u : ((b & 0x7FFFFFu) | 0x800000u);
      u32 Ee = z ? 0u : (((b >> 23) & 0xFFu) | 0x100u);
      lds[FW0 + kkL * WFS + row] = mb;
      lds[FW1 + kkL * WFS + row] =
          ((mb >> 3) & 0x111111u) | (Ee << 21) | (b & 0x80000000u);
    }
    __syncthreads();                    // features ready

    // ---- compute: 8 products per kk, two u64 MADs each ----
    const int xo = ty * TM;
    const int wo = tx * TN;
#pragma unroll 2
    for (int kk = 0; kk < KC; ++kk) {
      u32 xw0[TM], xv5[TM];
#pragma unroll
      for (int r = 0; r < TM; ++r) {             // merges into ds_load_b64
        xw0[r] = lds[FX0 + kk * XFS + xo + r];
        xv5[r] = lds[FX1 + kk * XFS + xo + r];
      }
      u32 wm[TN], ww1[TN];
#pragma unroll
      for (int c = 0; c < TN; ++c) {             // merges into ds_load_b128
        wm[c]  = lds[FW0 + kk * WFS + wo + c];
        ww1[c] = lds[FW1 + kk * WFS + wo + c];
      }
      u32 xU[TM]; int xE[TM];
#pragma unroll
      for (int r = 0; r < TM; ++r) {
        xU[r] = xw0[r] & 0x1FFFFFu;
        xE[r] = (int)((xw0[r] >> 21) & 0x3FFu);
      }
      u32 wQ[TN]; int wE[TN];
#pragma unroll
      for (int c = 0; c < TN; ++c) {
        wQ[c] = ww1[c] & 0x1FFFFFu;
        wE[c] = (int)((ww1[c] >> 21) & 0x3FFu);
      }
#pragma unroll
      for (int r = 0; r < TM; ++r)
#pragma unroll
        for (int c = 0; c < TN; ++c) {
          u64 mm = (u64)xU[r] * wm[c] + (u64)xv5[r] * wQ[c];
          int e = xE[r] + wE[c] - 639;            // == ea+eb-127 (or <0 if zero)
          e = e < 0 ? 0 : (e > 255 ? 255 : e);    // v_med3
          u32 t2  = (u32)(mm >> 47);              // 0..3 (bit47, bit48)
          u32 top = t2 < 1u ? t2 : 1u;            // leading-bit carry
          u32 sh  = 23u + top + (t2 >> 1);        // 23 / 24 / 25
          int e2  = e + (int)top;
          e2 = e2 > 255 ? 255 : e2;
          u32 mant = (u32)(mm >> sh) & 0x7FFFFFu;
          u32 fb = ((xw0[r] ^ ww1[c]) & 0x80000000u) | ((u32)e2 << 23) | mant;
          acc[r][c] += __uint_as_float(fb);       // zero-class -> +-0.0f
        }
    }
    __syncthreads();                    // features consumed; buffers reusable
    rb ^= 1;
  }

  // ---- epilogue: bias add, store ----
  const int m0 = blockIdx.y * BM + ty * TM;
  const int n0 = blockIdx.x * BN + tx * TN;
#pragma unroll
  for (int r = 0; r < TM; ++r)
#pragma unroll
    for (int c = 0; c < TN; ++c)
      outg[(size_t)(m0 + r) * N_ + n0 + c] = acc[r][c] + bg[n0 + c];
}

extern "C" void kernel_launch(void* const* d_in, const int* in_sizes, int n_in,
                              void* d_out, int out_size, void* d_ws, size_t ws_size,
                              hipStream_t stream) {
  (void)n_in; (void)out_size; (void)d_ws; (void)ws_size;
  const float* x = (const float*)d_in[0];   // [M,K] f32
  const float* w = (const float*)d_in[1];   // [N,K] f32
  const float* b = (const float*)d_in[2];   // [N]   f32
  float* out = (float*)d_out;               // [M,N] f32

  const int N = in_sizes[2];                // 512
  const int K = in_sizes[1] / N;            // 512
  const int M = in_sizes[0] / K;            // 256

  dim3 grid((unsigned)(N / BN), (unsigned)(M / BM));  // 8 x 8 = 64 blocks
  axc_linear_kernel<<<grid, 256, 0, stream>>>(x, w, b, out, K, N);
}